// MyHuberLoss_41360535061163
// MI455X (gfx1250) — compile-verified
//
#include <hip/hip_runtime.h>

// Memory-bound streaming reduction for pseudo-Huber loss.
// 224 MB read once -> non-temporal B128 loads, wave32 shuffle reduction,
// deterministic two-pass (partials in d_ws, then single-block finish).

typedef __attribute__((ext_vector_type(4))) float v4f;
typedef __attribute__((ext_vector_type(2))) float v2f;

#define NBLOCKS 2048
#define NTHREADS 256

// S(d) = sqrt(1 + (d/0.4)^2) = sqrt(fma(d*d, 6.25, 1))
__device__ __forceinline__ float huber_s(float d) {
    return __builtin_sqrtf(fmaf(d * d, 6.25f, 1.0f));
}

// Per-row loss with the delta^2 / alpha / beta / (-1) constants folded:
// 0.16*S(p5-t) + 0.0096 * sum_j [S(pj-t) + S(pj-p5)] - 0.256
__device__ __forceinline__ float row_loss(float p0, float p1, float p2,
                                          float p3, float p4, float p5,
                                          float t) {
    float s0 = huber_s(p5 - t);
    float sab = huber_s(p0 - t)  + huber_s(p1 - t)  + huber_s(p2 - t)
              + huber_s(p3 - t)  + huber_s(p4 - t)
              + huber_s(p0 - p5) + huber_s(p1 - p5) + huber_s(p2 - p5)
              + huber_s(p3 - p5) + huber_s(p4 - p5);
    return fmaf(0.0096f, sab, fmaf(0.16f, s0, -0.256f));
}

__device__ __forceinline__ float block_reduce(float acc) {
    // wave32 tree reduction (lowered to LDS-unit permutes on gfx1250)
    #pragma unroll
    for (int off = 16; off > 0; off >>= 1)
        acc += __shfl_down(acc, off, 32);

    __shared__ float smem[32];
    int lane = threadIdx.x & 31;
    int wave = threadIdx.x >> 5;
    if (lane == 0) smem[wave] = acc;
    __syncthreads();

    int nwaves = blockDim.x >> 5;
    float v = 0.0f;
    if (wave == 0) {
        v = (lane < nwaves) ? smem[lane] : 0.0f;
        #pragma unroll
        for (int off = 16; off > 0; off >>= 1)
            v += __shfl_down(v, off, 32);
    }
    return v;  // valid in wave 0, lane 0
}

__global__ void huber_partial_kernel(const float* __restrict__ out6,
                                     const float* __restrict__ lab,
                                     float* __restrict__ partial,
                                     long long pairs, long long rows) {
    float acc = 0.0f;
    const long long stride = (long long)gridDim.x * blockDim.x;
    long long p = (long long)blockIdx.x * blockDim.x + threadIdx.x;

    for (; p < pairs; p += stride) {
        const v4f* base = (const v4f*)(out6 + p * 12);  // 48B per row pair, 16B aligned
        // prefetch next grid-stride iteration's cacheline (global_prefetch_b8)
        if (p + stride < pairs)
            __builtin_prefetch((const void*)(out6 + (p + stride) * 12), 0, 0);

        // single-pass stream: non-temporal so we don't thrash the 192MB L2
        v4f a = __builtin_nontemporal_load(base + 0);
        v4f b = __builtin_nontemporal_load(base + 1);
        v4f c = __builtin_nontemporal_load(base + 2);
        v2f t = __builtin_nontemporal_load((const v2f*)(lab + p * 2));

        // row 2p   : pred = a.x a.y a.z a.w b.x, final = b.y, target = t.x
        acc += row_loss(a.x, a.y, a.z, a.w, b.x, b.y, t.x);
        // row 2p+1 : pred = b.z b.w c.x c.y c.z, final = c.w, target = t.y
        acc += row_loss(b.z, b.w, c.x, c.y, c.z, c.w, t.y);
    }

    // odd-row tail (not hit for n = 8M, kept for generality)
    if ((rows & 1LL) &&
        ((long long)blockIdx.x * blockDim.x + threadIdx.x) == 0) {
        long long r = rows - 1;
        const float* pr = out6 + r * 6;
        acc += row_loss(pr[0], pr[1], pr[2], pr[3], pr[4], pr[5], lab[r]);
    }

    float v = block_reduce(acc);
    if (threadIdx.x == 0) partial[blockIdx.x] = v;
}

__global__ void huber_final_kernel(const float* __restrict__ partial,
                                   int npart, float* __restrict__ out,
                                   float inv_n) {
    float acc = 0.0f;
    for (int i = threadIdx.x; i < npart; i += blockDim.x)
        acc += partial[i];
    float v = block_reduce(acc);
    if (threadIdx.x == 0) out[0] = v * inv_n;
}

extern "C" void kernel_launch(void* const* d_in, const int* in_sizes, int n_in,
                              void* d_out, int out_size, void* d_ws, size_t ws_size,
                              hipStream_t stream) {
    const float* out6 = (const float*)d_in[0];  // (n, 6) float32
    const float* lab  = (const float*)d_in[1];  // (n, 1) float32
    float* out        = (float*)d_out;          // scalar float32
    float* partial    = (float*)d_ws;           // NBLOCKS floats of scratch

    long long rows  = (long long)in_sizes[0] / 6;
    long long pairs = rows / 2;
    float inv_n = 1.0f / (float)rows;

    huber_partial_kernel<<<NBLOCKS, NTHREADS, 0, stream>>>(out6, lab, partial,
                                                           pairs, rows);
    huber_final_kernel<<<1, NTHREADS, 0, stream>>>(partial, NBLOCKS, out, inv_n);
}